// TritonLigerLMHeadCE_34634616275194
// MI455X (gfx1250) — compile-verified
//
#include <hip/hip_runtime.h>
#include <math.h>
#include <stdint.h>

// ---------------------------------------------------------------------------
// Fused LM-head cross-entropy for MI455X (gfx1250, wave32).
//   loss = mean_t [ logsumexp_v(x[t]·W[v]) - x[t]·W[target[t]] ]
// HBM-bound: 576 MB mandatory traffic (~25 us @ 23.3 TB/s) vs 1.07 TFLOP of
// f32 WMMA -> stream W once via the Tensor Data Mover (async, TENSORcnt),
// keep x + the W slab L2-resident (192 MB L2), never materialize logits.
// ---------------------------------------------------------------------------

#define HDIM 4096
#define VDIM 32000
#define BT   4096
#define IGNORE_INDEX (-100)

constexpr int TM  = 32;           // tokens per workgroup tile
constexpr int TN  = 256;          // vocab cols per workgroup tile
constexpr int TK  = 32;           // K-chunk staged through LDS
constexpr int NC  = HDIM / TK;    // 128 K-chunks
constexpr int NVB = VDIM / TN;    // 125 vocab blocks
constexpr int PAD = 36;           // LDS row stride (floats): TDM pads 32 DW + 4 DW

typedef float v2f __attribute__((ext_vector_type(2)));
typedef float v8f __attribute__((ext_vector_type(8)));
typedef unsigned int v4u __attribute__((ext_vector_type(4)));
typedef int v4i __attribute__((ext_vector_type(4)));
typedef int v8i __attribute__((ext_vector_type(8)));

// --- TDM descriptor helpers (D# per CDNA5 ISA) --------
// Group0: [1:0]=count=1, [63:32]=lds_addr, [120:64]=global_addr, [127:126]=type=2
__device__ __forceinline__ v4u tdm_g0(uint32_t lds_byte, const void* gptr) {
    const uint64_t ga = (uint64_t)(uintptr_t)gptr;
    v4u g0;
    g0.x = 1u;                                            // count=1, user mode
    g0.y = lds_byte;
    g0.z = (uint32_t)ga;
    g0.w = (uint32_t)((ga >> 32) & 0x01FFFFFFu) | (2u << 30);  // addr[56:32] | type=2
    return g0;
}
// Group1: data_size=4B, pad_enable, pad_interval=32 DW (enc 4), pad_amount=4 DW
// (enc 3) -> LDS row stride 36 floats. 2D tile: tile_dim0=TK, tile_dim1=rows.
__device__ __forceinline__ v8i tdm_g1(int tensor_rows, int tile_rows) {
    v8i g1;
    g1[0] = (2 << 16) | (1 << 20) | (4 << 22) | (3 << 25);
    g1[1] = (int)((HDIM & 0xFFFF) << 16);                         // tensor_dim0 lo16
    g1[2] = (int)(((HDIM >> 16) & 0xFFFF) | ((tensor_rows & 0xFFFF) << 16));
    g1[3] = (int)(((tensor_rows >> 16) & 0xFFFF) | (TK << 16));   // tile_dim0=32
    g1[4] = tile_rows;                                            // tile_dim1, tile_dim2=0
    g1[5] = HDIM;                                                 // tensor_dim0_stride lo32
    g1[6] = 0;
    g1[7] = 0;
    return g1;
}

__device__ __forceinline__ void tdm_issue(v4u g0, v8i g1) {
    const v4i z4 = {0, 0, 0, 0};
#if defined(__clang_major__) && (__clang_major__ >= 23)
    const v8i z8 = {0, 0, 0, 0, 0, 0, 0, 0};
    __builtin_amdgcn_tensor_load_to_lds(g0, g1, z4, z4, z8, 0);
#else
    __builtin_amdgcn_tensor_load_to_lds(g0, g1, z4, z4, 0);
#endif
}

// ---------------------------------------------------------------------------
// Kernel 1: tiled GEMM (x @ W^T) with on-the-fly row max / sum-exp.
// TDM double-buffered LDS staging; f32 WMMA 16x16x4 compute.
// ---------------------------------------------------------------------------
__global__ __launch_bounds__(256) void ce_partial_kernel(
    const float* __restrict__ x, const float* __restrict__ W,
    float2* __restrict__ partials)
{
    __shared__ float xs[2][TM * PAD];   // x tiles (double buffered)
    __shared__ float wt[2][TN * PAD];   // W tiles (double buffered)
    __shared__ float sm[4][TM];
    __shared__ float ssum[4][TM];

    const int tid  = threadIdx.x;
    const int wave = tid >> 5;
    const int lane = tid & 31;
    const int half = lane >> 4;
    const int ln   = lane & 15;

    const int tok0 = blockIdx.x * TM;   // fast grid index -> W slab reused in L2
    const int v0   = blockIdx.y * TN;

    const int wm  = wave & 1;           // 16-row half of the token tile
    const int wg  = wave >> 1;          // n-group 0..3 (64 cols each)
    const int cb0 = wg * 64;

    const uint32_t xs_lds[2] = { (uint32_t)(uintptr_t)&xs[0][0],
                                 (uint32_t)(uintptr_t)&xs[1][0] };
    const uint32_t wt_lds[2] = { (uint32_t)(uintptr_t)&wt[0][0],
                                 (uint32_t)(uintptr_t)&wt[1][0] };

    v8f acc0 = {}, acc1 = {}, acc2 = {}, acc3 = {};

    // Prologue: wave 0 issues the TDM pair for chunk 0 into buffer 0.
    if (wave == 0) {
        tdm_issue(tdm_g0(wt_lds[0], W + (size_t)v0 * HDIM), tdm_g1(VDIM, TN));
        tdm_issue(tdm_g0(xs_lds[0], x + (size_t)tok0 * HDIM), tdm_g1(BT, TM));
    }

    for (int c = 0; c < NC; ++c) {
        const int cur = c & 1;
        // Issue next chunk into the other buffer (released at end of c-1).
        if (wave == 0) {
            if (c + 1 < NC) {
                const int h0n = (c + 1) * TK;
                tdm_issue(tdm_g0(wt_lds[cur ^ 1], W + (size_t)v0 * HDIM + h0n),
                          tdm_g1(VDIM, TN));
                tdm_issue(tdm_g0(xs_lds[cur ^ 1], x + (size_t)tok0 * HDIM + h0n),
                          tdm_g1(BT, TM));
                __builtin_amdgcn_s_wait_tensorcnt((short)2);  // pair(c) complete
            } else {
                __builtin_amdgcn_s_wait_tensorcnt((short)0);
            }
        }
        __syncthreads();   // publish buffer `cur` to all waves

        // Contiguous v2f fragment loads (ds_load_b64, even VGPR pairs):
        // A 16x4 f32 layout: VGPR0=K{0|2}, VGPR1=K{1|3} by lane half.
        const float* ap  = &xs[cur][(wm * 16 + ln) * PAD + half * 2];
        const float* bp0 = &wt[cur][(cb0 + ln) * PAD + half * 2];
        const float* bp1 = bp0 + 16 * PAD;
        const float* bp2 = bp0 + 32 * PAD;
        const float* bp3 = bp0 + 48 * PAD;

#pragma unroll
        for (int kk = 0; kk < TK; kk += 4) {
            const v2f a  = *(const v2f*)(ap + kk);
            const v2f b0 = *(const v2f*)(bp0 + kk);
            acc0 = __builtin_amdgcn_wmma_f32_16x16x4_f32(false, a, false, b0,
                                                         (short)0, acc0, false, false);
            const v2f b1 = *(const v2f*)(bp1 + kk);
            acc1 = __builtin_amdgcn_wmma_f32_16x16x4_f32(false, a, false, b1,
                                                         (short)0, acc1, false, false);
            const v2f b2 = *(const v2f*)(bp2 + kk);
            acc2 = __builtin_amdgcn_wmma_f32_16x16x4_f32(false, a, false, b2,
                                                         (short)0, acc2, false, false);
            const v2f b3 = *(const v2f*)(bp3 + kk);
            acc3 = __builtin_amdgcn_wmma_f32_16x16x4_f32(false, a, false, b3,
                                                         (short)0, acc3, false, false);
        }
        __syncthreads();   // all waves done reading `cur` before it is refilled
    }

    // Per-row (m, s) over this wave's 64 columns.
    // C/D layout: VGPR r, half h, lane ln -> row wm*16 + h*8 + r, col cb0+16j+ln
#pragma unroll
    for (int r = 0; r < 8; ++r) {
        float m = fmaxf(fmaxf(acc0[r], acc1[r]), fmaxf(acc2[r], acc3[r]));
        m = fmaxf(m, __shfl_xor(m, 1));   // masks <= 8 stay within the 16-lane half
        m = fmaxf(m, __shfl_xor(m, 2));
        m = fmaxf(m, __shfl_xor(m, 4));
        m = fmaxf(m, __shfl_xor(m, 8));
        float s = expf(acc0[r] - m) + expf(acc1[r] - m) +
                  expf(acc2[r] - m) + expf(acc3[r] - m);
        s += __shfl_xor(s, 1);
        s += __shfl_xor(s, 2);
        s += __shfl_xor(s, 4);
        s += __shfl_xor(s, 8);
        if (ln == 0) {
            const int row = wm * 16 + half * 8 + r;
            sm[wg][row]   = m;
            ssum[wg][row] = s;
        }
    }
    __syncthreads();

    if (tid < TM) {
        const float m0 = sm[0][tid], m1 = sm[1][tid], m2 = sm[2][tid], m3 = sm[3][tid];
        const float M  = fmaxf(fmaxf(m0, m1), fmaxf(m2, m3));
        const float S  = ssum[0][tid] * expf(m0 - M) + ssum[1][tid] * expf(m1 - M) +
                         ssum[2][tid] * expf(m2 - M) + ssum[3][tid] * expf(m3 - M);
        partials[(size_t)(tok0 + tid) * NVB + blockIdx.y] = make_float2(M, S);
    }
}

// ---------------------------------------------------------------------------
// Kernel 2: target logit  tgt_logit[t] = x[t] . W[target[t]]
// ---------------------------------------------------------------------------
__global__ __launch_bounds__(256) void tgt_logit_kernel(
    const float* __restrict__ x, const float* __restrict__ W,
    const int* __restrict__ target, float* __restrict__ tgt_logit)
{
    __shared__ float red[256];
    const int t   = blockIdx.x;
    const int tgt = target[t];
    if (tgt == IGNORE_INDEX) {
        if (threadIdx.x == 0) tgt_logit[t] = 0.0f;
        return;
    }
    const float* xr = x + (size_t)t * HDIM;
    const float* wr = W + (size_t)tgt * HDIM;
    float s = 0.0f;
    for (int h = threadIdx.x; h < HDIM; h += 256) s = fmaf(xr[h], wr[h], s);
    red[threadIdx.x] = s;
    __syncthreads();
    for (int o = 128; o > 0; o >>= 1) {
        if (threadIdx.x < o) red[threadIdx.x] += red[threadIdx.x + o];
        __syncthreads();
    }
    if (threadIdx.x == 0) tgt_logit[t] = red[0];
}

// ---------------------------------------------------------------------------
// Kernel 3: merge 125 partial (m,s) per token -> per-token loss
// ---------------------------------------------------------------------------
__global__ __launch_bounds__(256) void ce_merge_kernel(
    const float2* __restrict__ partials, const float* __restrict__ tgt_logit,
    const int* __restrict__ target, float* __restrict__ loss_tok)
{
    const int t = blockIdx.x * blockDim.x + threadIdx.x;
    if (t >= BT) return;
    const float2* p = partials + (size_t)t * NVB;
    float M = -INFINITY, S = 0.0f;
    for (int j = 0; j < NVB; ++j) {
        const float2 v = p[j];
        if (v.x > M) { S = S * expf(M - v.x) + v.y; M = v.x; }
        else         { S += v.y * expf(v.x - M); }
    }
    const float lse = M + logf(S);
    loss_tok[t] = (target[t] != IGNORE_INDEX) ? (lse - tgt_logit[t]) : 0.0f;
}

// ---------------------------------------------------------------------------
// Kernel 4: mean over valid tokens -> scalar loss
// ---------------------------------------------------------------------------
__global__ __launch_bounds__(256) void ce_final_kernel(
    const float* __restrict__ loss_tok, const int* __restrict__ target,
    float* __restrict__ out)
{
    __shared__ float rs[256];
    __shared__ int   rc[256];
    float s = 0.0f;
    int   c = 0;
    for (int t = threadIdx.x; t < BT; t += 256) {
        s += loss_tok[t];
        c += (target[t] != IGNORE_INDEX) ? 1 : 0;
    }
    rs[threadIdx.x] = s;
    rc[threadIdx.x] = c;
    __syncthreads();
    for (int o = 128; o > 0; o >>= 1) {
        if (threadIdx.x < o) {
            rs[threadIdx.x] += rs[threadIdx.x + o];
            rc[threadIdx.x] += rc[threadIdx.x + o];
        }
        __syncthreads();
    }
    if (threadIdx.x == 0) out[0] = rs[0] / (float)(rc[0] > 1 ? rc[0] : 1);
}

// ---------------------------------------------------------------------------
// Launch. Inputs (setup_inputs order): 0=x f32 [BT,H], 1=target int, 2=W f32 [V,H]
// ---------------------------------------------------------------------------
extern "C" void kernel_launch(void* const* d_in, const int* in_sizes, int n_in,
                              void* d_out, int out_size, void* d_ws, size_t ws_size,
                              hipStream_t stream)
{
    (void)in_sizes; (void)n_in; (void)out_size; (void)ws_size;

    const float* x      = (const float*)d_in[0];
    const int*   target = (const int*)d_in[1];
    const float* W      = (const float*)d_in[2];
    float*       out    = (float*)d_out;

    char* ws = (char*)d_ws;
    float2* partials  = (float2*)ws;                                   // 4 MB
    float*  tgt_logit = (float*)(ws + (size_t)BT * NVB * sizeof(float2));
    float*  loss_tok  = tgt_logit + BT;

    dim3 grid1(BT / TM, NVB);   // token tiles fast -> W slab L2 reuse
    ce_partial_kernel<<<grid1, 256, 0, stream>>>(x, W, partials);
    tgt_logit_kernel<<<BT, 256, 0, stream>>>(x, W, target, tgt_logit);
    ce_merge_kernel<<<(BT + 255) / 256, 256, 0, stream>>>(partials, tgt_logit, target, loss_tok);
    ce_final_kernel<<<1, 256, 0, stream>>>(loss_tok, target, out);
}